// GraphSAGE_69793218560134
// MI455X (gfx1250) — compile-verified
//
#include <hip/hip_runtime.h>
#include <hip/hip_bf16.h>

typedef __attribute__((ext_vector_type(2))) float v2f;
typedef __attribute__((ext_vector_type(8))) float v8f;

#define DIM 128

// ---------------------------------------------------------------------------
// Zero a float buffer (grid-stride).
// ---------------------------------------------------------------------------
__global__ void zero_f32(float* __restrict__ p, size_t n) {
    size_t i = (size_t)blockIdx.x * blockDim.x + threadIdx.x;
    size_t stride = (size_t)gridDim.x * blockDim.x;
    for (; i < n; i += stride) p[i] = 0.0f;
}

// ---------------------------------------------------------------------------
// Degree count: cnt[dst[e]] += 1  (float atomics, one thread per edge)
// ---------------------------------------------------------------------------
__global__ void sage_degree(const long long* __restrict__ dst,
                            float* __restrict__ cnt, int E) {
    int i = blockIdx.x * blockDim.x + threadIdx.x;
    int stride = gridDim.x * blockDim.x;
    for (; i < E; i += stride) {
        int d = (int)dst[i];
        __hip_atomic_fetch_add(&cnt[d], 1.0f,
                               __ATOMIC_RELAXED, __HIP_MEMORY_SCOPE_AGENT);
    }
}

// ---------------------------------------------------------------------------
// Edge aggregation: agg[dst] += feat[src].  One wave per edge; each lane
// handles 4 contiguous features (float4 gather + 4 f32 atomic adds).
// Traffic stays in L2 (feature table = 51 MB < 192 MB L2).
// ---------------------------------------------------------------------------
__global__ void sage_aggregate(const float* __restrict__ feat,
                               const long long* __restrict__ src,
                               const long long* __restrict__ dst,
                               float* __restrict__ agg, int E) {
    int wid   = (blockIdx.x * blockDim.x + threadIdx.x) >> 5;
    int nwave = (gridDim.x * blockDim.x) >> 5;
    int lane  = threadIdx.x & 31;
    for (int e = wid; e < E; e += nwave) {
        int s = (int)src[e];
        int d = (int)dst[e];
        const float4 v = *(const float4*)(feat + (size_t)s * DIM + lane * 4);
        float* o = agg + (size_t)d * DIM + lane * 4;
        __hip_atomic_fetch_add(o + 0, v.x, __ATOMIC_RELAXED, __HIP_MEMORY_SCOPE_AGENT);
        __hip_atomic_fetch_add(o + 1, v.y, __ATOMIC_RELAXED, __HIP_MEMORY_SCOPE_AGENT);
        __hip_atomic_fetch_add(o + 2, v.z, __ATOMIC_RELAXED, __HIP_MEMORY_SCOPE_AGENT);
        __hip_atomic_fetch_add(o + 3, v.w, __ATOMIC_RELAXED, __HIP_MEMORY_SCOPE_AGENT);
    }
}

// ---------------------------------------------------------------------------
// Fused SAGE linear:  out = (agg * 1/max(cnt,1)) @ w_l^T + b + xin @ w_r^T
// (optional ReLU).  Full-f32 path via V_WMMA_F32_16X16X4_F32.
//
// Each BLOCK owns one 16-wide column strip n0 = (blockIdx.x & 7) * 16 and
// stages w_l/w_r rows n0..n0+15 (K=128) into LDS in [k][n] layout; the
// compiler hoists the per-wave B-fragments into VGPRs across the m-loop.
// A-fragments are prefetched in groups of 8 (8 loads in flight) to avoid
// the single-load s_wait_loadcnt 0 serialization per WMMA step.
// A-fragment: lane reads float2 A[m0 + (lane&15)][4*ks + 2*(lane>>4)].
// C/D tile: c[r] holds (M = r + 8*(lane>>4), N = lane&15).
// Requires N % 16 == 0 (N = 100000 here).
// ---------------------------------------------------------------------------
__global__ void __launch_bounds__(256)
sage_gemm(const float* __restrict__ agg,
          const float* __restrict__ cnt,
          const float* __restrict__ xin,
          const float* __restrict__ w_l,
          const float* __restrict__ b,
          const float* __restrict__ w_r,
          float* __restrict__ out,
          int N, int relu) {
    __shared__ float wl_lds[DIM * 16];   // 8 KB
    __shared__ float wr_lds[DIM * 16];   // 8 KB

    const int lane = threadIdx.x & 31;
    const int half = lane >> 4;      // 0: K pair {0,1}, 1: K pair {2,3}
    const int r    = lane & 15;
    const int waveInBlock = threadIdx.x >> 5;   // 0..7

    const int nTile = blockIdx.x & 7;           // 8 column tiles (DIM=128)
    const int n0 = nTile * 16;

    // Cooperative stage of this block's weight strips into LDS ([k][n]).
    for (int idx = threadIdx.x; idx < DIM * 16; idx += 256) {
        const int k = idx >> 4;
        const int n = idx & 15;
        wl_lds[k * 16 + n] = w_l[(size_t)(n0 + n) * DIM + k];
        wr_lds[k * 16 + n] = w_r[(size_t)(n0 + n) * DIM + k];
    }
    __syncthreads();

    // Per-wave B fragments (conflict-free ds reads; hoisted to VGPRs).
    v2f wlf[32], wrf[32];
#pragma unroll
    for (int ks = 0; ks < 32; ++ks) {
        const int kk = ks * 4 + half * 2;
        wlf[ks].x = wl_lds[kk * 16 + r];
        wlf[ks].y = wl_lds[(kk + 1) * 16 + r];
        wrf[ks].x = wr_lds[kk * 16 + r];
        wrf[ks].y = wr_lds[(kk + 1) * 16 + r];
    }

    const float bias = b[n0 + r];               // column bias for this lane

    const int mTiles  = N >> 4;
    const int mStart  = (blockIdx.x >> 3) * 8 + waveInBlock;
    const int mStride = (gridDim.x >> 3) * 8;

    for (int mt = mStart; mt < mTiles; mt += mStride) {
        const int m0   = mt * 16;
        const int arow = m0 + r;
        const float sc = 1.0f / fmaxf(cnt[arow], 1.0f);   // mean scaling
        const float* aggRow = agg + (size_t)arow * DIM + half * 2;
        const float* xRow   = xin + (size_t)arow * DIM + half * 2;

        v8f c = {};
        // --- K-loop over agg (scaled) @ w_l^T, 4 groups of 8 K-steps ---
#pragma unroll
        for (int g = 0; g < 4; ++g) {
            v2f a[8];
#pragma unroll
            for (int j = 0; j < 8; ++j)           // 8 loads in flight
                a[j] = *(const v2f*)(aggRow + (g * 8 + j) * 4);
#pragma unroll
            for (int j = 0; j < 8; ++j) {
                a[j].x *= sc; a[j].y *= sc;
                c = __builtin_amdgcn_wmma_f32_16x16x4_f32(
                        false, a[j], false, wlf[g * 8 + j], (short)0, c,
                        false, false);
            }
        }
        // --- K-loop over xin @ w_r^T ---
#pragma unroll
        for (int g = 0; g < 4; ++g) {
            v2f a[8];
#pragma unroll
            for (int j = 0; j < 8; ++j)
                a[j] = *(const v2f*)(xRow + (g * 8 + j) * 4);
#pragma unroll
            for (int j = 0; j < 8; ++j) {
                c = __builtin_amdgcn_wmma_f32_16x16x4_f32(
                        false, a[j], false, wrf[g * 8 + j], (short)0, c,
                        false, false);
            }
        }

        // Epilogue: bias (+ ReLU), scatter C tile per ISA layout.
#pragma unroll
        for (int rr = 0; rr < 8; ++rr) {
            const int row = m0 + rr + half * 8;
            float v = c[rr] + bias;
            if (relu) v = fmaxf(v, 0.0f);
            out[(size_t)row * DIM + (n0 + r)] = v;
        }
    }
}

// ---------------------------------------------------------------------------
// Host launch: 2-layer GraphSAGE forward.
// Inputs: x, edge_index(int64 [2,E]), w1_l, b1, w1_r, w2_l, b2, w2_r.
// ---------------------------------------------------------------------------
extern "C" void kernel_launch(void* const* d_in, const int* in_sizes, int n_in,
                              void* d_out, int out_size, void* d_ws, size_t ws_size,
                              hipStream_t stream) {
    const float*     x    = (const float*)d_in[0];
    const long long* ei   = (const long long*)d_in[1];
    const float*     w1_l = (const float*)d_in[2];
    const float*     b1   = (const float*)d_in[3];
    const float*     w1_r = (const float*)d_in[4];
    const float*     w2_l = (const float*)d_in[5];
    const float*     b2   = (const float*)d_in[6];
    const float*     w2_r = (const float*)d_in[7];
    float* out = (float*)d_out;

    const int N = in_sizes[0] / DIM;
    const int E = in_sizes[1] / 2;
    const long long* src = ei;        // edge_index[0]
    const long long* dst = ei + E;    // edge_index[1]

    const size_t featElems = (size_t)N * DIM;
    float* agg = (float*)d_ws;            // [N,128]
    float* h   = agg + featElems;         // [N,128]
    float* cnt = h + featElems;           // [N]

    // --- degree counts (shared by both layers) ---
    zero_f32<<<256, 256, 0, stream>>>(cnt, (size_t)N);
    sage_degree<<<2048, 256, 0, stream>>>(dst, cnt, E);

    // --- layer 1 ---
    zero_f32<<<2048, 256, 0, stream>>>(agg, featElems);
    sage_aggregate<<<8192, 256, 0, stream>>>(x, src, dst, agg, E);
    sage_gemm<<<640, 256, 0, stream>>>(agg, cnt, x, w1_l, b1, w1_r, h, N, 1);

    // --- layer 2 ---
    zero_f32<<<2048, 256, 0, stream>>>(agg, featElems);
    sage_aggregate<<<8192, 256, 0, stream>>>(h, src, dst, agg, E);
    sage_gemm<<<640, 256, 0, stream>>>(agg, cnt, h, w2_l, b2, w2_r, out, N, 0);
}